// GATEncoder_58050777972867
// MI455X (gfx1250) — compile-verified
//
#include <hip/hip_runtime.h>
#include <hip/hip_bf16.h>

typedef __attribute__((ext_vector_type(2))) float v2f;
typedef __attribute__((ext_vector_type(8))) float v8f;

#define HC 128     // H*C
#define HEADS 4
#define CH 32
#define NEG_SLOPE 0.2f

// ---------------- order-preserving float<->uint for atomic max ---------------
__device__ __forceinline__ unsigned f2ord(float f) {
    unsigned u = __float_as_uint(f);
    return (u & 0x80000000u) ? ~u : (u | 0x80000000u);
}
__device__ __forceinline__ float ord2f(unsigned u) {
    return (u & 0x80000000u) ? __uint_as_float(u & 0x7FFFFFFFu)
                             : __uint_as_float(~u);
}
#define ORD_NEG_INF 0x007FFFFFu   // f2ord(-inf)

__device__ __forceinline__ float lrelu(float v) { return v > 0.f ? v : NEG_SLOPE * v; }

// ---------------- fp32 WMMA GEMM: D[M,128] = A[M,128] * B[128,128] ----------
// One wave computes one 16x16 tile; 8 waves/block cover the 8 column tiles of
// a 16-row strip. M must be a multiple of 16 (50000 = 3125*16). EXEC is all-1s
// everywhere a WMMA executes.
__global__ __launch_bounds__(256) void gemm_f32_wmma(
    const float* __restrict__ A, const float* __restrict__ B,
    float* __restrict__ D, int M)
{
    const int wave = threadIdx.x >> 5;   // 0..7 -> N tile
    const int lane = threadIdx.x & 31;
    const int mt   = blockIdx.x;         // M tile
    const int m15  = lane & 15;
    const int hi   = lane >> 4;          // 0/1
    const int row  = mt * 16 + m15;
    const int col  = wave * 16 + m15;

    v8f acc;
    #pragma unroll
    for (int r = 0; r < 8; ++r) acc[r] = 0.f;

    #pragma unroll 4
    for (int k0 = 0; k0 < HC; k0 += 4) {
        // A 16x4 fp32 layout: VGPR0 = A[m][k0+2*hi], VGPR1 = A[m][k0+2*hi+1]
        const float* ap = A + (long)row * HC + k0 + 2 * hi;
        v2f a; a.x = ap[0]; a.y = ap[1];
        // B 4x16 fp32 layout: VGPR0 = B[k0+2*hi][n], VGPR1 = B[k0+2*hi+1][n]
        const float* bp = B + (long)(k0 + 2 * hi) * HC + col;
        v2f b; b.x = bp[0]; b.y = bp[HC];
        acc = __builtin_amdgcn_wmma_f32_16x16x4_f32(
            false, a, false, b, (short)0, acc, false, false);
    }
    // C/D 16x16 f32 layout: VGPR r holds rows r (lanes 0-15) and r+8 (16-31)
    #pragma unroll
    for (int r = 0; r < 8; ++r)
        D[(long)(mt * 16 + r + 8 * hi) * HC + col] = acc[r];
}

// ---------------- per-node attention logits ---------------------------------
__global__ void node_alphas(const float* __restrict__ h,
                            const float* __restrict__ a_src,
                            const float* __restrict__ a_dst,
                            float* __restrict__ as, float* __restrict__ ad,
                            int NH)
{
    int i = blockIdx.x * blockDim.x + threadIdx.x;   // over N*HEADS
    if (i >= NH) return;
    int n = i >> 2, hd = i & 3;
    const float* hp  = h + (long)n * HC + hd * CH;
    const float* asp = a_src + hd * CH;
    const float* adp = a_dst + hd * CH;
    float s1 = 0.f, s2 = 0.f;
    #pragma unroll
    for (int c = 0; c < CH; ++c) { float v = hp[c]; s1 += v * asp[c]; s2 += v * adp[c]; }
    as[i] = s1; ad[i] = s2;
}

// ---------------- fills ------------------------------------------------------
__global__ void fill_f32(float* p, float v, int n) {
    int i = blockIdx.x * blockDim.x + threadIdx.x;
    if (i < n) p[i] = v;
}
__global__ void fill_u32(unsigned* p, unsigned v, int n) {
    int i = blockIdx.x * blockDim.x + threadIdx.x;
    if (i < n) p[i] = v;
}

// ---------------- edge pass 1: segment max ----------------------------------
__global__ void edge_max(const int* __restrict__ ei, int E, int EL,
                         const float* __restrict__ as, const float* __restrict__ ad,
                         unsigned* __restrict__ m)
{
    int i = blockIdx.x * blockDim.x + threadIdx.x;
    if (i >= EL) return;
    int s, d;
    if (i < E) { s = ei[i]; d = ei[E + i]; } else { s = d = i - E; }
    const float4 a = *(const float4*)(as + (long)s * 4);
    const float4 b = *(const float4*)(ad + (long)d * 4);
    unsigned* mp = m + (long)d * 4;
    atomicMax(mp + 0, f2ord(lrelu(a.x + b.x)));
    atomicMax(mp + 1, f2ord(lrelu(a.y + b.y)));
    atomicMax(mp + 2, f2ord(lrelu(a.z + b.z)));
    atomicMax(mp + 3, f2ord(lrelu(a.w + b.w)));
}

// ---------------- edge pass 2: segment sum of exp ---------------------------
__global__ void edge_sum(const int* __restrict__ ei, int E, int EL,
                         const float* __restrict__ as, const float* __restrict__ ad,
                         const unsigned* __restrict__ m, float* __restrict__ ss)
{
    int i = blockIdx.x * blockDim.x + threadIdx.x;
    if (i >= EL) return;
    int s, d;
    if (i < E) { s = ei[i]; d = ei[E + i]; } else { s = d = i - E; }
    const float4 a = *(const float4*)(as + (long)s * 4);
    const float4 b = *(const float4*)(ad + (long)d * 4);
    const unsigned* mp = m + (long)d * 4;
    float* sp = ss + (long)d * 4;
    atomicAdd(sp + 0, __expf(lrelu(a.x + b.x) - ord2f(mp[0])));
    atomicAdd(sp + 1, __expf(lrelu(a.y + b.y) - ord2f(mp[1])));
    atomicAdd(sp + 2, __expf(lrelu(a.z + b.z) - ord2f(mp[2])));
    atomicAdd(sp + 3, __expf(lrelu(a.w + b.w) - ord2f(mp[3])));
}

// ---------------- edge pass 3: weighted scatter-add (1 wave / edge) ---------
__global__ __launch_bounds__(256) void edge_agg(
    const int* __restrict__ ei, int E, int EL,
    const float* __restrict__ h,
    const float* __restrict__ as, const float* __restrict__ ad,
    const unsigned* __restrict__ m, const float* __restrict__ ss,
    float* __restrict__ out)
{
    int e = blockIdx.x * 8 + (threadIdx.x >> 5);
    if (e >= EL) return;
    int lane = threadIdx.x & 31;
    int s, d;
    if (e < E) { s = ei[e]; d = ei[E + e]; } else { s = d = e - E; }
    int hd = lane >> 3;                          // head of this lane's 4 channels
    float ev    = lrelu(as[(long)s * 4 + hd] + ad[(long)d * 4 + hd]);
    float alpha = __expf(ev - ord2f(m[(long)d * 4 + hd])) / ss[(long)d * 4 + hd];
    const float4 hv = *(const float4*)(h + (long)s * HC + lane * 4);
    float* op = out + (long)d * HC + lane * 4;
    atomicAdd(op + 0, hv.x * alpha);
    atomicAdd(op + 1, hv.y * alpha);
    atomicAdd(op + 2, hv.z * alpha);
    atomicAdd(op + 3, hv.w * alpha);
}

// ---------------- bias + ELU (in place) -------------------------------------
__global__ void bias_elu(float* __restrict__ y, const float* __restrict__ b, int total)
{
    int i = blockIdx.x * blockDim.x + threadIdx.x;
    if (i >= total) return;
    float v = y[i] + b[i & (HC - 1)];
    y[i] = v > 0.f ? v : __expf(v) - 1.f;
}

// ---------------------------------------------------------------------------
extern "C" void kernel_launch(void* const* d_in, const int* in_sizes, int n_in,
                              void* d_out, int out_size, void* d_ws, size_t ws_size,
                              hipStream_t stream)
{
    const float* x      = (const float*)d_in[0];
    const int*   ei     = (const int*)d_in[1];
    const float* W1     = (const float*)d_in[2];
    const float* a_src1 = (const float*)d_in[3];
    const float* a_dst1 = (const float*)d_in[4];
    const float* b1     = (const float*)d_in[5];
    const float* W2     = (const float*)d_in[6];
    const float* a_src2 = (const float*)d_in[7];
    const float* a_dst2 = (const float*)d_in[8];
    const float* b2     = (const float*)d_in[9];

    const int N  = in_sizes[0] / HC;     // 50000
    const int E  = in_sizes[1] / 2;      // 1,600,000
    const int EL = E + N;                // edges + self loops
    const int NH = N * HEADS;
    const int NF = N * HC;

    // workspace layout (bytes)
    char* ws = (char*)d_ws;
    float*    hbuf = (float*)(ws);                          // N*HC f32 (h1, then h2)
    float*    out1 = (float*)(ws + (size_t)NF * 4);         // N*HC f32
    float*    as   = (float*)(ws + (size_t)NF * 8);         // N*H
    float*    ad   = (float*)(ws + (size_t)NF * 8 + (size_t)NH * 4);
    unsigned* mArr = (unsigned*)(ws + (size_t)NF * 8 + (size_t)NH * 8);
    float*    sArr = (float*)(ws + (size_t)NF * 8 + (size_t)NH * 12);

    const int TB = 256;
    const int gN  = (NH + TB - 1) / TB;
    const int gNF = (NF + TB - 1) / TB;
    const int gE  = (EL + TB - 1) / TB;
    const int gEW = (EL + 7) / 8;        // wave-per-edge kernels
    float* outF = (float*)d_out;

    // =================== layer 1 ===================
    gemm_f32_wmma<<<N / 16, 256, 0, stream>>>(x, W1, hbuf, N);
    node_alphas<<<gN, TB, 0, stream>>>(hbuf, a_src1, a_dst1, as, ad, NH);
    fill_u32<<<gN, TB, 0, stream>>>(mArr, ORD_NEG_INF, NH);
    fill_f32<<<gN, TB, 0, stream>>>(sArr, 0.f, NH);
    fill_f32<<<gNF, TB, 0, stream>>>(out1, 0.f, NF);
    edge_max<<<gE, TB, 0, stream>>>(ei, E, EL, as, ad, mArr);
    edge_sum<<<gE, TB, 0, stream>>>(ei, E, EL, as, ad, mArr, sArr);
    edge_agg<<<gEW, TB, 0, stream>>>(ei, E, EL, hbuf, as, ad, mArr, sArr, out1);
    bias_elu<<<gNF, TB, 0, stream>>>(out1, b1, NF);

    // =================== layer 2 ===================
    gemm_f32_wmma<<<N / 16, 256, 0, stream>>>(out1, W2, hbuf, N);
    node_alphas<<<gN, TB, 0, stream>>>(hbuf, a_src2, a_dst2, as, ad, NH);
    fill_u32<<<gN, TB, 0, stream>>>(mArr, ORD_NEG_INF, NH);
    fill_f32<<<gN, TB, 0, stream>>>(sArr, 0.f, NH);
    fill_f32<<<gNF, TB, 0, stream>>>(outF, 0.f, NF);
    edge_max<<<gE, TB, 0, stream>>>(ei, E, EL, as, ad, mArr);
    edge_sum<<<gE, TB, 0, stream>>>(ei, E, EL, as, ad, mArr, sArr);
    edge_agg<<<gEW, TB, 0, stream>>>(ei, E, EL, hbuf, as, ad, mArr, sArr, outF);
    bias_elu<<<gNF, TB, 0, stream>>>(outF, b2, NF);
}